// SemanticContrastiveLoss_10849087389943
// MI455X (gfx1250) — compile-verified
//
#include <hip/hip_runtime.h>
#include <hip/hip_bf16.h>

typedef float v2f __attribute__((ext_vector_type(2)));
typedef float v4f __attribute__((ext_vector_type(4)));
typedef float v8f __attribute__((ext_vector_type(8)));
typedef int   v4i __attribute__((ext_vector_type(4)));

#define BS    4096
#define D     256
#define NLAB  64
#define ROWS  16            // rows per workgroup
#define SROW  4100          // padded LDS row stride (floats) to break 4096%64 bank alias
#define NT    256           // threads per block (8 waves of 32)
#define NCT   (BS / 16)     // 256 column tiles per row stripe

// ---------------------------------------------------------------------------
// Main fused kernel: one WG per 16-row stripe.
//   Phase 1: 16x4096 score block via V_WMMA_F32_16X16X4_F32 -> LDS (diag masked)
//            4-way column-tile register blocking: 1 A load + 4 B loads feed
//            4 WMMAs per k-step (A fragment shared across tiles).
//   Phase 2: per-row max, then clipped-logit logsumexp + weighted sums
//   Output : ws[2*bid] = sum over rows of mean_log_prob (has_pos rows only)
//            ws[2*bid+1] = count of has_pos rows
// ---------------------------------------------------------------------------
extern "C" __global__ void __launch_bounds__(NT, 1)
scl_main(const float* __restrict__ F, const int* __restrict__ L,
         const float* __restrict__ S, float* __restrict__ ws)
{
    extern __shared__ float smem[];
    float* scores = smem;                  // ROWS * SROW floats (~256 KB)
    float* redV   = smem + ROWS * SROW;    // 16 per-row mean_log_prob
    float* redC   = redV + ROWS;           // 16 per-row has_pos flags

    const int r0    = blockIdx.x * ROWS;
    const int tid   = threadIdx.x;
    const int w     = tid >> 5;
    const int lane  = tid & 31;
    const int lhalf = lane >> 4;           // 0: K pair (0,1)-half, 1: (2,3)-half
    const int l15   = lane & 15;

    // ---------------- Phase 1: WMMA GEMM into LDS ----------------
    // A fragment (16x4 f32): lane l15 holds row r0+l15, K pair at k + 2*lhalf.
    const float* Abase = F + (size_t)(r0 + l15) * D + (lhalf << 1);

    // Wave w owns tiles {w + 8t}. Group as 8 passes of 4 tiles:
    // pass g handles tiles ctb, ctb+8, ctb+16, ctb+24 with ctb = w + 32g.
    // Tile ctb+8j starts at column ctb*16 + 128j -> B offset 128*D*j floats,
    // foldable into the 24-bit load immediate.
    for (int g = 0; g < 8; ++g) {
        const int ctb = w + (g << 5);
        const float* Bbase = F + (size_t)((ctb << 4) + l15) * D + (lhalf << 1);

        v8f acc0 = {}, acc1 = {}, acc2 = {}, acc3 = {};
        for (int k = 0; k < D; k += 4) {
            v2f a  = *(const v2f*)(Abase + k);
            v2f b0 = *(const v2f*)(Bbase + k);
            v2f b1 = *(const v2f*)(Bbase + 128 * D + k);
            v2f b2 = *(const v2f*)(Bbase + 256 * D + k);
            v2f b3 = *(const v2f*)(Bbase + 384 * D + k);
            acc0 = __builtin_amdgcn_wmma_f32_16x16x4_f32(false, a, false, b0,
                                                         (short)0, acc0, false, false);
            acc1 = __builtin_amdgcn_wmma_f32_16x16x4_f32(false, a, false, b1,
                                                         (short)0, acc1, false, false);
            acc2 = __builtin_amdgcn_wmma_f32_16x16x4_f32(false, a, false, b2,
                                                         (short)0, acc2, false, false);
            acc3 = __builtin_amdgcn_wmma_f32_16x16x4_f32(false, a, false, b3,
                                                         (short)0, acc3, false, false);
        }

        // C layout: VGPR r -> M = r + 8*lhalf, N = l15. Apply diag mask on store.
#pragma unroll
        for (int j = 0; j < 4; ++j) {
            const v8f acc = (j == 0) ? acc0 : (j == 1) ? acc1 : (j == 2) ? acc2 : acc3;
            const int gcol = ((ctb + 8 * j) << 4) + l15;
#pragma unroll
            for (int r = 0; r < 8; ++r) {
                const int m = r + (lhalf << 3);
                float v = acc[r];
                if (gcol == r0 + m) v = -1.0e9f;
                scores[m * SROW + gcol] = v;
            }
        }
    }
    __syncthreads();

    // ---------------- Phase 2: per-row reductions ----------------
    // Wave w owns rows 2w and 2w+1.
    for (int rr = 0; rr < 2; ++rr) {
        const int m = (w << 1) | rr;
        const int i = r0 + m;
        const float* srow = scores + m * SROW;

        // Pass A: row max (includes the -1e9 diagonal, which never wins).
        float mx = -3.4e38f;
        for (int t = lane; t < BS / 4; t += 32) {
            v4f x = ((const v4f*)srow)[t];
            mx = fmaxf(mx, fmaxf(fmaxf(x[0], x[1]), fmaxf(x[2], x[3])));
        }
#pragma unroll
        for (int off = 16; off > 0; off >>= 1)
            mx = fmaxf(mx, __shfl_xor(mx, off, 32));

        // Pass B: Z = sum exp(clip), S1 = sum w*logit, W = sum w, C = pos count.
        const int li = L[i];
        const float* srT = S + li * NLAB;
        float Z = 0.f, S1 = 0.f, W = 0.f, C = 0.f;
        for (int t = lane; t < BS / 4; t += 32) {
            v4f x  = ((const v4f*)srow)[t];
            v4i lj = ((const v4i*)L)[t];
#pragma unroll
            for (int e = 0; e < 4; ++e) {
                const int j = (t << 2) + e;
                float lg = x[e] - mx;
                lg = fminf(fmaxf(lg, -50.f), 50.f);
                Z += expf(lg);                       // all j contribute (incl. diag @ -50)
                const float s = srT[lj[e]];
                if (j != i) {
                    const bool organ = (s > 0.f) & (s < 1.f);
                    const bool exact = (s == 1.f);
                    if (organ | exact) {
                        const float wj = organ ? s * 4.0f : 2.0f; // FINE_W / ORGAN_W
                        S1 += wj * lg;
                        W  += wj;
                        C  += 1.0f;
                    }
                }
            }
        }
#pragma unroll
        for (int off = 16; off > 0; off >>= 1) {
            Z  += __shfl_xor(Z,  off, 32);
            S1 += __shfl_xor(S1, off, 32);
            W  += __shfl_xor(W,  off, 32);
            C  += __shfl_xor(C,  off, 32);
        }
        if (lane == 0) {
            const float mlp = (S1 - W * logf(Z + 1e-8f)) / fmaxf(W, 1e-8f);
            redV[m] = (C > 0.f) ? mlp : 0.f;
            redC[m] = (C > 0.f) ? 1.f : 0.f;
        }
    }
    __syncthreads();

    if (tid == 0) {
        float sv = 0.f, sc = 0.f;
#pragma unroll
        for (int m = 0; m < ROWS; ++m) { sv += redV[m]; sc += redC[m]; }
        ws[2 * blockIdx.x]     = sv;
        ws[2 * blockIdx.x + 1] = sc;
    }
}

// ---------------------------------------------------------------------------
// Final scalar: loss = -sum(mean_log_prob over has_pos) / max(n_pos, 1)
// ---------------------------------------------------------------------------
extern "C" __global__ void scl_finalize(const float* __restrict__ ws,
                                        float* __restrict__ out)
{
    if (threadIdx.x == 0) {
        float sv = 0.f, sn = 0.f;
        for (int b = 0; b < BS / ROWS; ++b) {
            sv += ws[2 * b];
            sn += ws[2 * b + 1];
        }
        out[0] = -sv / fmaxf(sn, 1.0f);
    }
}

extern "C" void kernel_launch(void* const* d_in, const int* in_sizes, int n_in,
                              void* d_out, int out_size, void* d_ws, size_t ws_size,
                              hipStream_t stream)
{
    const float* F = (const float*)d_in[0];   // features (4096, 256) f32
    const int*   L = (const int*)d_in[1];     // labels   (4096,)     i32
    const float* S = (const float*)d_in[2];   // sim_table (64, 64)   f32
    float* out = (float*)d_out;
    float* ws  = (float*)d_ws;                // 512 floats of per-WG partials

    const size_t smem_bytes = (size_t)(ROWS * SROW + 2 * ROWS) * sizeof(float);

    scl_main<<<BS / ROWS, NT, smem_bytes, stream>>>(F, L, S, ws);
    scl_finalize<<<1, 64, 0, stream>>>(ws, out);
}